// HeteroGATLayer_47124381171980
// MI455X (gfx1250) — compile-verified
//
#include <hip/hip_runtime.h>
#include <math.h>

#define NP 200000
#define NF 50000
#define EN 1000000
#define EH 1000000
#define ET 500000

typedef float v2f __attribute__((ext_vector_type(2)));
typedef float v8f __attribute__((ext_vector_type(8)));

__device__ __forceinline__ float wredsum(float v) {
#pragma unroll
  for (int off = 16; off; off >>= 1) v += __shfl_xor(v, off, 32);
  return v;
}

// Monotonic float atomic-max: non-negative via signed int max, negative via unsigned min.
__device__ __forceinline__ void atomicMaxF(float* a, float v) {
  if (v >= 0.0f) atomicMax((int*)a, __float_as_int(v));
  else           atomicMin((unsigned int*)a, __float_as_uint(v));
}

// ---------------- small prep kernels ----------------

// out[r] = sum_c W[r*128+c] * a[c], r < rows (rows <= 128)
__global__ void rowdot128(const float* __restrict__ W, const float* __restrict__ a,
                          float* __restrict__ out, int rows) {
  int r = blockIdx.x * blockDim.x + threadIdx.x;
  if (r < rows) {
    float s = 0.0f;
    for (int c = 0; c < 128; ++c) s += W[r * 128 + c] * a[c];
    out[r] = s;
  }
}

// out[row] = x[row,:] . v   (wave per row, 128-dim)
__global__ void matvec128(const float* __restrict__ x, const float* __restrict__ v,
                          float* __restrict__ out, int n) {
  int row = blockIdx.x * (blockDim.x >> 5) + (threadIdx.x >> 5);
  int lane = threadIdx.x & 31;
  if (row >= n) return;
  const float4* xp = (const float4*)(x + (size_t)row * 128);
  const float4* vp = (const float4*)v;
  float4 a = xp[lane], b = vp[lane];
  float s = wredsum(a.x * b.x + a.y * b.y + a.z * b.z + a.w * b.w);
  if (lane == 0) out[row] = s;
}

__global__ void fillf(float* __restrict__ p, float v, long long n) {
  long long i = (long long)blockIdx.x * blockDim.x + threadIdx.x;
  if (i < n) p[i] = v;
}

__global__ void bcast_bias(float* __restrict__ out, const float* __restrict__ bias,
                           long long total) {
  long long i = (long long)blockIdx.x * blockDim.x + threadIdx.x;
  if (i < total) out[i] = bias[i & 127];
}

// ---------------- WMMA f32 GEMM: C[nrows,128] (op) A[nrows,128] @ B[128,128] ---------
// MODE 0: C = A*B
// MODE 1: C += A*B
// MODE 2: C += A*B + bias[col]
// MODE 3: C = gelu(A*B + bias[col] + resid[row,col])   (exact GELU)
template <int MODE>
__global__ void gemm128(const float* __restrict__ A, const float* __restrict__ B,
                        float* __restrict__ C, const float* __restrict__ bias,
                        const float* __restrict__ resid, int nrows) {
  const int lane = threadIdx.x & 31;
  const int wave = threadIdx.x >> 5;
  const int half = lane >> 4;       // K-half selector per ISA A/B layout
  const int r    = lane & 15;       // M (for A) / N (for B,C,D)
  const int row0 = (blockIdx.x * (blockDim.x >> 5) + wave) << 4;
  if (row0 >= nrows) return;        // wave-uniform: EXEC stays all-1s for WMMA

  v8f acc[8] = {};
  const float* arow = A + (size_t)(row0 + r) * 128;

  for (int k0 = 0; k0 < 128; k0 += 4) {
    v2f a;
    a.x = arow[k0 + 2 * half + 0];  // VGPR0: K = k0 + 2*half
    a.y = arow[k0 + 2 * half + 1];  // VGPR1: K = k0 + 2*half + 1
#pragma unroll
    for (int n = 0; n < 8; ++n) {
      const int col = (n << 4) + r;
      v2f b;
      b.x = B[(size_t)(k0 + 2 * half + 0) * 128 + col];
      b.y = B[(size_t)(k0 + 2 * half + 1) * 128 + col];
      acc[n] = __builtin_amdgcn_wmma_f32_16x16x4_f32(
          false, a, false, b, (short)0, acc[n], false, false);
    }
  }

#pragma unroll
  for (int n = 0; n < 8; ++n) {
    const int col = (n << 4) + r;
#pragma unroll
    for (int m = 0; m < 8; ++m) {
      const int row = row0 + m + (half << 3);   // lanes 16-31 hold M = 8..15
      const size_t idx = (size_t)row * 128 + col;
      float v = acc[n][m];
      if (MODE == 0)      C[idx] = v;
      else if (MODE == 1) C[idx] += v;
      else if (MODE == 2) C[idx] += v + bias[col];
      else {
        float t = v + bias[col] + resid[idx];
        C[idx] = 0.5f * t * (1.0f + erff(t * 0.70710678118654752f));
      }
    }
  }
}

// ---------------- edge kernels ----------------

// wave per edge: logit = leaky_relu(als[src] + ald[dst] + ea[e,:].ve, 0.2); atomic max into mx[dst]
__global__ void edge_logits(const float* __restrict__ ea, const float* __restrict__ ve,
                            const float* __restrict__ als, const float* __restrict__ ald,
                            const int* __restrict__ src, const int* __restrict__ dst,
                            float* __restrict__ logits, float* __restrict__ mx, int E) {
  int e = blockIdx.x * (blockDim.x >> 5) + (threadIdx.x >> 5);
  int lane = threadIdx.x & 31;
  if (e >= E) return;
  float p = ea[(size_t)e * 32 + lane] * ve[lane];
  p = wredsum(p);
  if (lane == 0) {
    float l = als[src[e]] + ald[dst[e]] + p;
    l = l > 0.0f ? l : 0.2f * l;
    logits[e] = l;
    atomicMaxF(&mx[dst[e]], l);
  }
}

// thread per edge: w = exp(logit - mx[dst]); overwrite logits with w; denom[dst] += w
__global__ void edge_exp(const int* __restrict__ dst, float* __restrict__ logits,
                         const float* __restrict__ mx, float* __restrict__ denom, int E) {
  int e = blockIdx.x * blockDim.x + threadIdx.x;
  if (e >= E) return;
  int d = dst[e];
  float m = mx[d];
  if (!isfinite(m)) m = 0.0f;
  float w = expf(logits[e] - m);
  logits[e] = w;
  atomicAdd(&denom[d], w);
}

// wave per edge: out[dst,:] += (w/denom) * xs[src,:]   (float4 per lane, atomics land in L2)
__global__ void edge_scatter(const int* __restrict__ src, const int* __restrict__ dst,
                             const float* __restrict__ expw, const float* __restrict__ denom,
                             const float* __restrict__ xs, float* __restrict__ out, int E) {
  int e = blockIdx.x * (blockDim.x >> 5) + (threadIdx.x >> 5);
  int lane = threadIdx.x & 31;
  if (e >= E) return;
  int s = src[e], d = dst[e];
  float a = expw[e] / fmaxf(denom[d], 1e-16f);
  float4 v = ((const float4*)(xs + (size_t)s * 128))[lane];
  float* o = out + (size_t)d * 128 + lane * 4;
  atomicAdd(o + 0, a * v.x);
  atomicAdd(o + 1, a * v.y);
  atomicAdd(o + 2, a * v.z);
  atomicAdd(o + 3, a * v.w);
}

// wave per edge: agg[dst,:] += x[src,:]; deg[dst] += 1
__global__ void sage_scatter(const int* __restrict__ src, const int* __restrict__ dst,
                             const float* __restrict__ x, float* __restrict__ agg,
                             float* __restrict__ deg, int E) {
  int e = blockIdx.x * (blockDim.x >> 5) + (threadIdx.x >> 5);
  int lane = threadIdx.x & 31;
  if (e >= E) return;
  int s = src[e], d = dst[e];
  if (lane == 0) atomicAdd(&deg[d], 1.0f);
  float4 v = ((const float4*)(x + (size_t)s * 128))[lane];
  float* o = agg + (size_t)d * 128 + lane * 4;
  atomicAdd(o + 0, v.x);
  atomicAdd(o + 1, v.y);
  atomicAdd(o + 2, v.z);
  atomicAdd(o + 3, v.w);
}

// agg[i] /= max(deg[i/128], 1)
__global__ void sage_norm(float* __restrict__ agg, const float* __restrict__ deg, long long n) {
  long long i = (long long)blockIdx.x * blockDim.x + threadIdx.x;
  if (i < n) agg[i] /= fmaxf(deg[i >> 7], 1.0f);
}

// wave per row layernorm over 128 channels
__global__ void layernorm128(const float* __restrict__ x, const float* __restrict__ g,
                             const float* __restrict__ b, float* __restrict__ y, int n) {
  int row = blockIdx.x * (blockDim.x >> 5) + (threadIdx.x >> 5);
  int lane = threadIdx.x & 31;
  if (row >= n) return;
  float4 v = ((const float4*)(x + (size_t)row * 128))[lane];
  float mu = wredsum(v.x + v.y + v.z + v.w) * (1.0f / 128.0f);
  float d0 = v.x - mu, d1 = v.y - mu, d2 = v.z - mu, d3 = v.w - mu;
  float var = wredsum(d0 * d0 + d1 * d1 + d2 * d2 + d3 * d3) * (1.0f / 128.0f);
  float rs = rsqrtf(var + 1e-5f);
  int c = lane * 4;
  float* yp = y + (size_t)row * 128 + c;
  yp[0] = d0 * rs * g[c + 0] + b[c + 0];
  yp[1] = d1 * rs * g[c + 1] + b[c + 1];
  yp[2] = d2 * rs * g[c + 2] + b[c + 2];
  yp[3] = d3 * rs * g[c + 3] + b[c + 3];
}

// ---------------- launch ----------------

extern "C" void kernel_launch(void* const* d_in, const int* in_sizes, int n_in,
                              void* d_out, int out_size, void* d_ws, size_t ws_size,
                              hipStream_t stream) {
  (void)in_sizes; (void)n_in; (void)out_size; (void)ws_size;

  const float* x_plume = (const float*)d_in[0];
  const float* x_fac   = (const float*)d_in[1];
  const float* ea_near = (const float*)d_in[2];
  const float* ea_hist = (const float*)d_in[3];
  const int*   nsrc    = (const int*)d_in[4];
  const int*   ndst    = (const int*)d_in[5];
  const int*   hsrc    = (const int*)d_in[6];
  const int*   hdst    = (const int*)d_in[7];
  const int*   tsrc    = (const int*)d_in[8];
  const int*   tdst    = (const int*)d_in[9];
  const float* Wn_src  = (const float*)d_in[10];
  const float* Wn_dst  = (const float*)d_in[11];
  const float* Wn_e    = (const float*)d_in[12];
  const float* an_src  = (const float*)d_in[13];
  const float* an_dst  = (const float*)d_in[14];
  const float* an_e    = (const float*)d_in[15];
  const float* bn      = (const float*)d_in[16];
  const float* Wh_src  = (const float*)d_in[17];
  const float* Wh_dst  = (const float*)d_in[18];
  const float* Wh_e    = (const float*)d_in[19];
  const float* ah_src  = (const float*)d_in[20];
  const float* ah_dst  = (const float*)d_in[21];
  const float* ah_e    = (const float*)d_in[22];
  const float* bh      = (const float*)d_in[23];
  const float* Ws_l    = (const float*)d_in[24];
  const float* bs_l    = (const float*)d_in[25];
  const float* Ws_r    = (const float*)d_in[26];
  const float* g_p     = (const float*)d_in[27];
  const float* beta_p  = (const float*)d_in[28];
  const float* g_f     = (const float*)d_in[29];
  const float* beta_f  = (const float*)d_in[30];
  const float* Wp_p    = (const float*)d_in[31];
  const float* bp_p    = (const float*)d_in[32];
  const float* Wp_f    = (const float*)d_in[33];
  const float* bp_f    = (const float*)d_in[34];

  float* w = (float*)d_ws;
  size_t o = 0;
  float* xs_near  = w + o; o += (size_t)NP * 128;
  float* xs_hist  = w + o; o += (size_t)NF * 128;
  float* out_p    = w + o; o += (size_t)NP * 128;
  float* out_f    = w + o; o += (size_t)NF * 128;
  float* sage     = w + o; o += (size_t)NP * 128;   // SAGE agg/mean, later reused as ln_p
  float* ln_f     = w + o; o += (size_t)NF * 128;
  float* als_near = w + o; o += NP;
  float* ald_near = w + o; o += NF;
  float* als_hist = w + o; o += NF;
  float* ald_hist = w + o; o += NP;
  float* mx_f     = w + o; o += NF;
  float* den_f    = w + o; o += NF;
  float* mx_p     = w + o; o += NP;
  float* den_p    = w + o; o += NP;
  float* deg      = w + o; o += NP;
  float* lg_near  = w + o; o += EN;
  float* lg_hist  = w + o; o += EH;
  float* vn_s     = w + o; o += 128;
  float* vn_d     = w + o; o += 128;
  float* vn_e     = w + o; o += 128;
  float* vh_s     = w + o; o += 128;
  float* vh_d     = w + o; o += 128;
  float* vh_e     = w + o; o += 128;

  float* yp = (float*)d_out;
  float* yf = yp + (size_t)NP * 128;

  // fold W@a into attention vectors
  rowdot128<<<1, 128, 0, stream>>>(Wn_src, an_src, vn_s, 128);
  rowdot128<<<1, 128, 0, stream>>>(Wn_dst, an_dst, vn_d, 128);
  rowdot128<<<1, 128, 0, stream>>>(Wn_e,   an_e,   vn_e, 32);
  rowdot128<<<1, 128, 0, stream>>>(Wh_src, ah_src, vh_s, 128);
  rowdot128<<<1, 128, 0, stream>>>(Wh_dst, ah_dst, vh_d, 128);
  rowdot128<<<1, 128, 0, stream>>>(Wh_e,   ah_e,   vh_e, 32);

  // node attention scalars
  matvec128<<<(NP + 7) / 8, 256, 0, stream>>>(x_plume, vn_s, als_near, NP);
  matvec128<<<(NF + 7) / 8, 256, 0, stream>>>(x_fac,   vn_d, ald_near, NF);
  matvec128<<<(NF + 7) / 8, 256, 0, stream>>>(x_fac,   vh_s, als_hist, NF);
  matvec128<<<(NP + 7) / 8, 256, 0, stream>>>(x_plume, vh_d, ald_hist, NP);

  // init reductions / outputs
  fillf<<<(NF + 255) / 256, 256, 0, stream>>>(mx_f, -INFINITY, NF);
  fillf<<<(NF + 255) / 256, 256, 0, stream>>>(den_f, 0.0f, NF);
  fillf<<<(NP + 255) / 256, 256, 0, stream>>>(mx_p, -INFINITY, NP);
  fillf<<<(NP + 255) / 256, 256, 0, stream>>>(den_p, 0.0f, NP);
  fillf<<<(NP + 255) / 256, 256, 0, stream>>>(deg, 0.0f, NP);
  fillf<<<((long long)NP * 128 + 255) / 256, 256, 0, stream>>>(sage, 0.0f, (long long)NP * 128);
  bcast_bias<<<((long long)NF * 128 + 255) / 256, 256, 0, stream>>>(out_f, bn, (long long)NF * 128);
  bcast_bias<<<((long long)NP * 128 + 255) / 256, 256, 0, stream>>>(out_p, bh, (long long)NP * 128);

  // message transforms (WMMA f32)
  gemm128<0><<<(NP + 127) / 128, 256, 0, stream>>>(x_plume, Wn_src, xs_near, nullptr, nullptr, NP);
  gemm128<0><<<(NF + 127) / 128, 256, 0, stream>>>(x_fac,   Wh_src, xs_hist, nullptr, nullptr, NF);

  // GAT near: plume -> facility
  edge_logits<<<(EN + 7) / 8, 256, 0, stream>>>(ea_near, vn_e, als_near, ald_near, nsrc, ndst, lg_near, mx_f, EN);
  edge_exp<<<(EN + 255) / 256, 256, 0, stream>>>(ndst, lg_near, mx_f, den_f, EN);
  edge_scatter<<<(EN + 7) / 8, 256, 0, stream>>>(nsrc, ndst, lg_near, den_f, xs_near, out_f, EN);

  // GAT history: facility -> plume
  edge_logits<<<(EH + 7) / 8, 256, 0, stream>>>(ea_hist, vh_e, als_hist, ald_hist, hsrc, hdst, lg_hist, mx_p, EH);
  edge_exp<<<(EH + 255) / 256, 256, 0, stream>>>(hdst, lg_hist, mx_p, den_p, EH);
  edge_scatter<<<(EH + 7) / 8, 256, 0, stream>>>(hsrc, hdst, lg_hist, den_p, xs_hist, out_p, EH);

  // SAGE: plume -> plume (mean aggr + root weight)
  sage_scatter<<<(ET + 7) / 8, 256, 0, stream>>>(tsrc, tdst, x_plume, sage, deg, ET);
  sage_norm<<<((long long)NP * 128 + 255) / 256, 256, 0, stream>>>(sage, deg, (long long)NP * 128);
  gemm128<2><<<(NP + 127) / 128, 256, 0, stream>>>(sage,    Ws_l, out_p, bs_l,   nullptr, NP);
  gemm128<1><<<(NP + 127) / 128, 256, 0, stream>>>(x_plume, Ws_r, out_p, nullptr, nullptr, NP);

  // LayerNorm -> proj -> +residual -> exact GELU (fused epilogue in WMMA GEMM)
  layernorm128<<<(NP + 7) / 8, 256, 0, stream>>>(out_p, g_p, beta_p, sage /*ln_p*/, NP);
  layernorm128<<<(NF + 7) / 8, 256, 0, stream>>>(out_f, g_f, beta_f, ln_f, NF);
  gemm128<3><<<(NP + 127) / 128, 256, 0, stream>>>(sage, Wp_p, yp, bp_p, x_plume, NP);
  gemm128<3><<<(NF + 127) / 128, 256, 0, stream>>>(ln_f, Wp_f, yf, bp_f, x_fac, NF);
}